// GIN_31224412242368
// MI455X (gfx1250) — compile-verified
//
#include <hip/hip_runtime.h>

typedef __attribute__((ext_vector_type(2))) float v2f;
typedef __attribute__((ext_vector_type(8))) float v8f;

#define B_    16
#define N_    128
#define NC    60     // N_NODE_CODE
#define NA    4      // N_NODE_ATTR
#define NI    64     // N_NODE_INFO
#define EAT   16     // N_EDGE_ATTR
#define EC    64     // N_EDGE_CODE
#define NIN   128    // NODE_IN
#define NOUT  60     // NODE_OUT
#define SLOPE 0.01f

__device__ __forceinline__ float lrelu(float v) { return v > 0.f ? v : v * SLOPE; }

__global__ __launch_bounds__(256) void gin_fused(
    const float* __restrict__ x,        const float* __restrict__ node_attr,
    const float* __restrict__ edge_attr,const float* __restrict__ adj,
    const float* __restrict__ We,       const float* __restrict__ be,
    const float* __restrict__ Wn,       const float* __restrict__ bn,
    float* __restrict__ out)
{
    __shared__ float s_send[N_ * EC];    // 128x64 send codes (32KB)
    __shared__ float s_recvb[16 * EC];   // recv + be for this i-tile (4KB)
    __shared__ float s_agg[16 * EC];     // edge_agg for this i-tile (4KB)
    __shared__ float s_adj[16 * N_];     // adj rows for this i-tile (8KB)

    const int b    = blockIdx.x >> 3;
    const int it   = blockIdx.x & 7;
    const int i0   = it * 16;

    const int tid  = threadIdx.x;
    const int wave = tid >> 5;           // 0..7 (wave32)
    const int lane = tid & 31;
    const int lh   = lane >> 4;          // lane half (K/M split per ISA layout)
    const int ll   = lane & 15;

    // ---- stage adj rows for this i-tile
    for (int t = tid; t < 16 * N_; t += 256) {
        int i = t >> 7, j = t & 127;
        s_adj[t] = adj[((b * N_) + i0 + i) * N_ + j];
    }

    // ==== Phase 1: send (all 128 nodes) and recv(+be) (16 i's) via WMMA f32 16x16x4, K=64
    // tiles 0..31: send (jt=t>>2, ct=t&3); tiles 32..35: recv (ct=t-32)
    for (int t = wave; t < 36; t += 8) {
        const bool is_recv = (t >= 32);
        const int  ct      = is_recv ? (t - 32) : (t & 3);
        const int  mrow0   = is_recv ? i0 : ((t >> 2) * 16);
        const int  wrow0   = is_recv ? EAT : (EAT + NI);   // We_r rows 16..79, We_s rows 80..143
        const int  node    = mrow0 + ll;
        const long nb      = (long)(b * N_ + node);
        v8f acc = {};
        #pragma unroll
        for (int k0 = 0; k0 < NI; k0 += 4) {
            const int ka = k0 + 2 * lh;
            v2f a, bb;
            if (k0 < NC) {                                   // compile-time per unrolled k0
                const float* p = x + nb * NC + ka;
                a.x = p[0]; a.y = p[1];
            } else {
                const float* p = node_attr + nb * NA + (ka - NC);
                a.x = p[0]; a.y = p[1];
            }
            const float* q = We + (wrow0 + ka) * EC + ct * 16 + ll;
            bb.x = q[0]; bb.y = q[EC];
            acc = __builtin_amdgcn_wmma_f32_16x16x4_f32(false, a, false, bb,
                                                        (short)0, acc, false, false);
        }
        #pragma unroll
        for (int r = 0; r < 8; ++r) {
            const int row = r + 8 * lh;
            const int col = ct * 16 + ll;
            if (is_recv) s_recvb[row * EC + col] = acc[r] + be[col];
            else         s_send[((t >> 2) * 16 + row) * EC + col] = acc[r];
        }
    }
    __syncthreads();

    // ==== Phase 2: edge MLP + leaky_relu + adj-weighted aggregation, fully fused.
    // Keep all of We_e (16x64) in registers: 4 c-tiles x 4 k-steps x v2f = 32 VGPRs.
    v2f bE[4][4];
    #pragma unroll
    for (int ct = 0; ct < 4; ++ct)
        #pragma unroll
        for (int kk = 0; kk < 4; ++kk) {
            const int k = 4 * kk + 2 * lh;
            const float* q = We + k * EC + ct * 16 + ll;
            bE[ct][kk].x = q[0]; bE[ct][kk].y = q[EC];
        }

    #pragma unroll
    for (int iw = 0; iw < 2; ++iw) {
        const int  il    = wave * 2 + iw;                  // 0..15 local i
        const int  i     = i0 + il;
        const long ebase = ((long)(b * N_ + i)) * N_ * EAT;
        float aggp[4] = {0.f, 0.f, 0.f, 0.f};
        float rb[4];
        #pragma unroll
        for (int ct = 0; ct < 4; ++ct) rb[ct] = s_recvb[il * EC + ct * 16 + ll];

        for (int jt = 0; jt < 8; ++jt) {
            v2f a[4];                                      // A tile: 16 j-rows x 16 k
            #pragma unroll
            for (int kk = 0; kk < 4; ++kk) {
                const int k = 4 * kk + 2 * lh;
                const float* p = edge_attr + ebase + (jt * 16 + ll) * EAT + k;
                a[kk].x = p[0]; a[kk].y = p[1];
            }
            #pragma unroll
            for (int ct = 0; ct < 4; ++ct) {
                v8f d = {};
                #pragma unroll
                for (int kk = 0; kk < 4; ++kk)
                    d = __builtin_amdgcn_wmma_f32_16x16x4_f32(false, a[kk], false, bE[ct][kk],
                                                              (short)0, d, false, false);
                // epilogue: d[r] = pre[j, c] with j = jt*16 + r + 8*lh, c = ct*16 + ll
                #pragma unroll
                for (int r = 0; r < 8; ++r) {
                    const int j = jt * 16 + r + 8 * lh;
                    float v = d[r] + rb[ct] + s_send[j * EC + ct * 16 + ll];
                    aggp[ct] = fmaf(lrelu(v), s_adj[il * N_ + j], aggp[ct]);
                }
            }
        }
        // combine the two lane-halves (each covered different j's for the same c)
        #pragma unroll
        for (int ct = 0; ct < 4; ++ct) {
            float v = aggp[ct] + __shfl_xor(aggp[ct], 16, 32);
            if (lh == 0) s_agg[il * EC + ct * 16 + ll] = v;
        }
    }
    __syncthreads();

    // ==== Phase 3: node MLP: (16 x 128) @ (128 x 60->64 padded), bias + leaky_relu.
    if (wave < 4) {
        const int ct = wave;
        const int o  = ct * 16 + ll;
        v8f acc = {};
        #pragma unroll
        for (int k0 = 0; k0 < NIN; k0 += 4) {
            const int ka = k0 + 2 * lh;
            v2f a, bb;
            if (k0 < NC) {                                 // node_info: x part
                const float* p = x + (long)(b * N_ + i0 + ll) * NC + ka;
                a.x = p[0]; a.y = p[1];
            } else if (k0 < NI) {                          // node_info: attr part
                const float* p = node_attr + (long)(b * N_ + i0 + ll) * NA + (ka - NC);
                a.x = p[0]; a.y = p[1];
            } else {                                       // edge_agg from LDS
                a.x = s_agg[ll * EC + (ka - NI)];
                a.y = s_agg[ll * EC + (ka - NI) + 1];
            }
            if (o < NOUT) {
                const float* q = Wn + ka * NOUT + o;
                bb.x = q[0]; bb.y = q[NOUT];
            } else {
                bb.x = 0.f; bb.y = 0.f;
            }
            acc = __builtin_amdgcn_wmma_f32_16x16x4_f32(false, a, false, bb,
                                                        (short)0, acc, false, false);
        }
        if (o < NOUT) {
            const float bv = bn[o];
            #pragma unroll
            for (int r = 0; r < 8; ++r) {
                const int node = i0 + r + 8 * lh;
                out[(long)(b * N_ + node) * NOUT + o] = lrelu(acc[r] + bv);
            }
        }
    }
}

extern "C" void kernel_launch(void* const* d_in, const int* in_sizes, int n_in,
                              void* d_out, int out_size, void* d_ws, size_t ws_size,
                              hipStream_t stream) {
    (void)in_sizes; (void)n_in; (void)d_ws; (void)ws_size; (void)out_size;
    const float* x         = (const float*)d_in[0];
    const float* node_attr = (const float*)d_in[1];
    const float* edge_attr = (const float*)d_in[2];
    const float* adj       = (const float*)d_in[3];
    const float* We        = (const float*)d_in[4];
    const float* be        = (const float*)d_in[5];
    const float* Wn        = (const float*)d_in[6];
    const float* bn        = (const float*)d_in[7];
    float* out             = (float*)d_out;

    dim3 grid(B_ * (N_ / 16));   // 128 blocks: one (b, i-tile)
    dim3 block(256);             // 8 waves (wave32)
    gin_fused<<<grid, block, 0, stream>>>(x, node_attr, edge_attr, adj, We, be, Wn, bn, out);
}